// UnderwaterEnhancer_65678639891040
// MI455X (gfx1250) — compile-verified
//
#include <hip/hip_runtime.h>
#include <hip/hip_bf16.h>
#include <math.h>

typedef __attribute__((ext_vector_type(2))) float v2f;
typedef __attribute__((ext_vector_type(8))) float v8f;

#define PLANE (1u << 20)          // 1024*1024
#define NPLANES 24                // 8*3
#define H 1024
#define W 1024

// ---- monotone float<->uint mapping for order-invariant atomic min/max ----
__device__ __forceinline__ unsigned ordF(float f) {
  unsigned u = __float_as_uint(f);
  return (u & 0x80000000u) ? ~u : (u | 0x80000000u);
}
__device__ __forceinline__ float unordF(unsigned o) {
  unsigned u = (o & 0x80000000u) ? (o ^ 0x80000000u) : ~o;
  return __uint_as_float(u);
}
__device__ __forceinline__ float clamp01(float v) {
  return fminf(fmaxf(v, 0.0f), 1.0f);
}

// ---- K1: per-plane partial sums (64 partials per plane, deterministic) ----
__global__ __launch_bounds__(256) void k_partials(const float* __restrict__ x,
                                                  float* __restrict__ partials) {
  int plane = blockIdx.y;         // 0..23
  int blk = blockIdx.x;           // 0..63
  const float* p = x + (size_t)plane * PLANE;
  size_t base = (size_t)blk * (256 * 64) + threadIdx.x;
  float s = 0.0f;
#pragma unroll 4
  for (int i = 0; i < 64; ++i) s += p[base + (size_t)i * 256];
  __shared__ float red[256];
  red[threadIdx.x] = s;
  __syncthreads();
  for (int off = 128; off > 0; off >>= 1) {
    if (threadIdx.x < (unsigned)off) red[threadIdx.x] += red[threadIdx.x + off];
    __syncthreads();
  }
  if (threadIdx.x == 0) partials[plane * 64 + blk] = red[0];
}

// ---- K1b: finish means -> gray-world scales; init global min/max ----
__global__ void k_scales(const float* __restrict__ partials,
                         float* __restrict__ scale_bc,
                         unsigned* __restrict__ gmm) {
  __shared__ float mean_rgb[NPLANES];
  int t = threadIdx.x;
  if (t < NPLANES) {
    float s = 0.0f;
    for (int i = 0; i < 64; ++i) s += partials[t * 64 + i];
    mean_rgb[t] = s * (1.0f / (1024.0f * 1024.0f));
  }
  __syncthreads();
  if (t < NPLANES) {
    int b = t / 3;
    float mg = (mean_rgb[b * 3 + 0] + mean_rgb[b * 3 + 1] + mean_rgb[b * 3 + 2]) * (1.0f / 3.0f);
    scale_bc[t] = mg / (mean_rgb[t] + 1e-6f);
  }
  if (t == 24) gmm[0] = 0xFFFFFFFFu;  // ordered min accumulator
  if (t == 25) gmm[1] = 0u;           // ordered max accumulator
}

// ---- K2: color-corrected channel min -> minch plane ----
__global__ __launch_bounds__(256) void k_minch(const float* __restrict__ x,
                                               const float* __restrict__ scale_bc,
                                               float* __restrict__ minch) {
  size_t idx = (size_t)blockIdx.x * 256 + threadIdx.x;  // 0..8M-1
  int b = (int)(idx >> 20);
  size_t hw = idx & (PLANE - 1);
  const float* px = x + ((size_t)b * 3) * PLANE + hw;
  float m = __builtin_inff();
#pragma unroll
  for (int c = 0; c < 3; ++c) {
    float v = clamp01(px[(size_t)c * PLANE] * scale_bc[b * 3 + c]);
    m = fminf(m, v);
  }
  minch[idx] = m;
}

// ---- K3: horizontal 15-wide min (pad +inf) ----
__global__ __launch_bounds__(256) void k_hmin(const float* __restrict__ minch,
                                              float* __restrict__ htmp) {
  __shared__ float s[256 + 14];
  int b = blockIdx.z, row = blockIdx.y, c0 = blockIdx.x * 256;
  const float* p = minch + ((size_t)b << 20) + (size_t)row * W;
  int t = threadIdx.x;
  for (int i = t; i < 270; i += 256) {
    int c = c0 - 7 + i;
    s[i] = (c >= 0 && c < W) ? p[c] : __builtin_inff();
  }
  __syncthreads();
  float m = s[t];
#pragma unroll
  for (int d = 1; d < 15; ++d) m = fminf(m, s[t + d]);
  htmp[((size_t)b << 20) + (size_t)row * W + c0 + t] = m;
}

// ---- K4: vertical 15-wide min (pad +inf) ----
__global__ __launch_bounds__(256) void k_vmin(const float* __restrict__ htmp,
                                              float* __restrict__ dark) {
  int b = blockIdx.z, r = blockIdx.y;
  int c = blockIdx.x * 256 + threadIdx.x;
  const float* p = htmp + ((size_t)b << 20);
  float m = __builtin_inff();
#pragma unroll
  for (int d = -7; d <= 7; ++d) {
    int rr = r + d;
    if (rr >= 0 && rr < H) m = fminf(m, p[(size_t)rr * W + c]);
  }
  dark[((size_t)b << 20) + (size_t)r * W + c] = m;
}

// ---- K6: fused color-correct + dehaze + 25x25 LCN (WMMA vertical sums) ----
#define LCN_TX 64
#define LCN_TY 32
#define LCN_LW 88   // TX + 24
#define LCN_LH 56   // TY + 24
__global__ __launch_bounds__(256) void k_lcn(const float* __restrict__ x,
                                             const float* __restrict__ dark,
                                             const float* __restrict__ scale_bc,
                                             const float* __restrict__ dsp,
                                             float* __restrict__ enh,
                                             unsigned* __restrict__ gmm) {
  __shared__ float tile[LCN_LH][LCN_LW + 1];   // dehazed values, zero-padded halo
  __shared__ float hs[LCN_LH][LCN_TX + 1];     // horizontal 25-sum of v
  __shared__ float hq[LCN_LH][LCN_TX + 1];     // horizontal 25-sum of v^2
  int plane = blockIdx.z;
  int b = plane / 3;
  int r0 = blockIdx.y * LCN_TY, c0 = blockIdx.x * LCN_TX;
  float ds = dsp[0];
  float sc = scale_bc[plane];
  const float* px = x + (size_t)plane * PLANE;
  const float* pd = dark + ((size_t)b << 20);
  int t = threadIdx.x;

  // Stage 1: fill tile with dehazed, color-corrected pixels (0 outside image)
  for (int i = t; i < LCN_LH * LCN_LW; i += 256) {
    int lr = i / LCN_LW, lc = i % LCN_LW;
    int gr = r0 - 12 + lr, gc = c0 - 12 + lc;
    float v2 = 0.0f;
    if (gr >= 0 && gr < H && gc >= 0 && gc < W) {
      size_t o = (size_t)gr * W + gc;
      float v = clamp01(px[o] * sc);
      float d = pd[o];
      v2 = clamp01((v - d) / (1.0f - ds * d + 1e-6f));
    }
    tile[lr][lc] = v2;
  }
  __syncthreads();

  // Stage 2: horizontal 25-sums of v and v^2 (56 rows x 64 cols)
  for (int i = t; i < LCN_LH * LCN_TX; i += 256) {
    int lr = i >> 6, j = i & 63;
    float s = 0.0f, q = 0.0f;
#pragma unroll
    for (int d = 0; d < 25; ++d) {
      float v = tile[lr][j + d];
      s += v;
      q += v * v;
    }
    hs[lr][j] = s;
    hq[lr][j] = q;
  }
  __syncthreads();

  // Stage 3: vertical 25-sums as banded matmul on the matrix pipe.
  // C(16x16) = A_band(16x40) x Hsum(40x16), K tiled by 4 -> 10 WMMA steps,
  // two accumulators (sum, sqsum). Wave w owns a 16x16 output sub-tile.
  int w = t >> 5, lane = t & 31;
  int hf = lane >> 4;       // lane half selects K pair (A) / row pair (B)
  int jj = lane & 15;       // A: row i of band; B: output column
  int wx = w & 3, wy = w >> 2;
  v8f cs = {};
  v8f cq = {};
#pragma unroll
  for (int kk = 0; kk < 10; ++kk) {
    int t0 = kk * 4 + 2 * hf;  // K index held in vgpr .x
    int t1 = t0 + 1;           // K index held in vgpr .y
    v2f a, bs, bq;
    // band: output row i sums window rows i..i+24 of the 40-row strip
    a.x = (t0 >= jj && t0 <= jj + 24) ? 1.0f : 0.0f;
    a.y = (t1 >= jj && t1 <= jj + 24) ? 1.0f : 0.0f;
    bs.x = hs[wy * 16 + t0][wx * 16 + jj];
    bs.y = hs[wy * 16 + t1][wx * 16 + jj];
    bq.x = hq[wy * 16 + t0][wx * 16 + jj];
    bq.y = hq[wy * 16 + t1][wx * 16 + jj];
    cs = __builtin_amdgcn_wmma_f32_16x16x4_f32(false, a, false, bs, (short)0, cs, false, false);
    cq = __builtin_amdgcn_wmma_f32_16x16x4_f32(false, a, false, bq, (short)0, cq, false, false);
  }

  // Stage 4: enh = (v - mean)/(std + 1e-4); track global min/max
  float lmin = __builtin_inff(), lmax = -__builtin_inff();
  float* pe = enh + (size_t)plane * PLANE;
#pragma unroll
  for (int r = 0; r < 8; ++r) {
    int orl = wy * 16 + r + 8 * hf;   // C layout: vgpr r holds rows r and r+8
    int ocl = wx * 16 + jj;
    float mean = cs[r] * (1.0f / 625.0f);
    float var = cq[r] * (1.0f / 625.0f) - mean * mean;
    var = fmaxf(var, 1e-4f);
    float st = sqrtf(var);
    float v = tile[orl + 12][ocl + 12];
    float e = (v - mean) / (st + 1e-4f);
    pe[(size_t)(r0 + orl) * W + (c0 + ocl)] = e;
    lmin = fminf(lmin, e);
    lmax = fmaxf(lmax, e);
  }
  __syncthreads();
  // block min/max reduce (reuse hs as scratch), then order-invariant atomics
  float* red = &hs[0][0];
  red[t] = lmin;
  red[256 + t] = lmax;
  __syncthreads();
  for (int off = 128; off > 0; off >>= 1) {
    if (t < off) {
      red[t] = fminf(red[t], red[t + off]);
      red[256 + t] = fmaxf(red[256 + t], red[256 + t + off]);
    }
    __syncthreads();
  }
  if (t == 0) {
    atomicMin(&gmm[0], ordF(red[0]));
    atomicMax(&gmm[1], ordF(red[256]));
  }
}

// ---- K6b: lo / inv scalars ----
__global__ void k_range(const unsigned* __restrict__ gmm, float* __restrict__ loinv) {
  float lo = unordF(gmm[0]);
  float hi = unordF(gmm[1]);
  loinv[0] = lo;
  loinv[1] = 1.0f / (hi - lo + 1e-4f);
}

// ---- K7: normalize + 3x3 unsharp + color scale ----
// blur(normalized) with zero-pad == inv*(sum_valid(enh) - cnt*lo)/9
__global__ __launch_bounds__(256) void k_final(const float* __restrict__ enh,
                                               const float* __restrict__ loinv,
                                               const float* __restrict__ ssp,
                                               const float* __restrict__ cscale,
                                               float* __restrict__ out) {
  int plane = blockIdx.z;
  int c = plane % 3;
  int r = blockIdx.y;
  int col = blockIdx.x * 256 + threadIdx.x;
  float lo = loinv[0], inv = loinv[1], ss = ssp[0], cs = cscale[c];
  const float* pe = enh + (size_t)plane * PLANE;
  float sumE = 0.0f;
  int cnt = 0;
  float ec = 0.0f;
#pragma unroll
  for (int dr = -1; dr <= 1; ++dr) {
    int rr = r + dr;
    if (rr < 0 || rr >= H) continue;
#pragma unroll
    for (int dc = -1; dc <= 1; ++dc) {
      int cc = col + dc;
      if (cc < 0 || cc >= W) continue;
      float e = pe[(size_t)rr * W + cc];
      sumE += e;
      cnt++;
      if (dr == 0 && dc == 0) ec = e;
    }
  }
  float n = (ec - lo) * inv;
  float blur = inv * (sumE - (float)cnt * lo) * (1.0f / 9.0f);
  float v = clamp01(n + ss * (n - blur));
  v = clamp01(v * cs);
  out[(size_t)plane * PLANE + (size_t)r * W + col] = v;
}

extern "C" void kernel_launch(void* const* d_in, const int* in_sizes, int n_in,
                              void* d_out, int out_size, void* d_ws, size_t ws_size,
                              hipStream_t stream) {
  const float* x = (const float*)d_in[0];        // [8,3,1024,1024]
  const float* cscale = (const float*)d_in[1];   // [3]
  const float* ss = (const float*)d_in[2];       // scalar
  const float* dsp = (const float*)d_in[3];      // scalar
  float* out = (float*)d_out;

  float* ws = (float*)d_ws;
  float* partials = ws;                           // 24*64
  float* scale_bc = ws + 1536;                    // 24
  unsigned* gmm = (unsigned*)(ws + 1560);         // 2 (ordered min/max)
  float* loinv = ws + 1562;                       // 2
  float* planeA = ws + 2048;                      // 8M floats (minch -> dark)
  float* planeB = planeA + (size_t)8 * PLANE;     // 8M floats (h-min tmp)
  float* enh = planeB + (size_t)8 * PLANE;        // 24M floats

  k_partials<<<dim3(64, NPLANES), 256, 0, stream>>>(x, partials);
  k_scales<<<1, 32, 0, stream>>>(partials, scale_bc, gmm);
  k_minch<<<32768, 256, 0, stream>>>(x, scale_bc, planeA);
  k_hmin<<<dim3(4, H, 8), 256, 0, stream>>>(planeA, planeB);
  k_vmin<<<dim3(4, H, 8), 256, 0, stream>>>(planeB, planeA);
  k_lcn<<<dim3(W / LCN_TX, H / LCN_TY, NPLANES), 256, 0, stream>>>(
      x, planeA, scale_bc, dsp, enh, gmm);
  k_range<<<1, 1, 0, stream>>>(gmm, loinv);
  k_final<<<dim3(4, H, NPLANES), 256, 0, stream>>>(enh, loinv, ss, cscale, out);
}